// lidar_DynamicEmbedder_90950227460852
// MI455X (gfx1250) — compile-verified
//
#include <hip/hip_runtime.h>
#include <hip/hip_bf16.h>

typedef __attribute__((ext_vector_type(2))) float v2f;
typedef __attribute__((ext_vector_type(8))) float v8f;

#define NPTS 150000
#define HW   262144      /* 512*512 */
#define WXD  512
#define EPSB 0.001f

/* ---- stats block layout (floats) ---- */
#define IDX_N    0
#define IDX_S1   1
#define IDX_S1Q  33
#define IDX_SC1  65
#define IDX_SH1  97
#define IDX_S2   129
#define IDX_S2Q  193
#define IDX_SC2  257
#define IDX_SH2  321

/* ---- workspace layout (bytes, 256-aligned) ---- */
static constexpr size_t SZ_STATS = 4096;
static constexpr size_t O_STATS  = 0;
static constexpr size_t O_VID    = O_STATS + SZ_STATS;
static constexpr size_t SZ_VID   = ((size_t)NPTS * 4 + 255) & ~(size_t)255;
static constexpr size_t O_CNT    = O_VID + SZ_VID;
static constexpr size_t SZ_CNT   = (size_t)HW * 4;
static constexpr size_t O_A      = O_CNT + SZ_CNT;     /* overlay region */
static constexpr size_t O_X1     = O_A;                /* N*32 f32 (y1 then x1) */
static constexpr size_t SZ_X1    = (size_t)NPTS * 32 * 4;
static constexpr size_t O_SXYZ   = O_X1 + SZ_X1;       /* HW*3 f32 */
static constexpr size_t SZ_SXYZ  = (size_t)HW * 3 * 4;
static constexpr size_t O_P1SUM  = O_SXYZ + SZ_SXYZ;   /* HW*32 f32 */
static constexpr size_t SZ_P1    = (size_t)HW * 32 * 4;
static constexpr size_t O_P1MAX  = O_P1SUM + SZ_P1;
static constexpr size_t END_A1   = O_P1MAX + SZ_P1;
static constexpr size_t O_P2MAX  = O_A;                /* overlays x1/sxyz/p1 (dead after K6) */
static constexpr size_t SZ_P2    = (size_t)HW * 64 * 4;
static constexpr size_t END_A2   = O_P2MAX + SZ_P2;
static constexpr size_t END_A    = (END_A1 > END_A2) ? END_A1 : END_A2;
static constexpr size_t O_Y2     = END_A;              /* N*64 f32 */
static constexpr size_t SZ_Y2    = (size_t)NPTS * 64 * 4;
static constexpr size_t O_P2SUM  = O_Y2 + SZ_Y2;
static constexpr size_t WS_NEEDED = O_P2SUM + SZ_P2;

/* ============ K1: voxelize, cnt, sum_xyz, valid count ============ */
__global__ __launch_bounds__(256) void k1_voxelize(
    const float* __restrict__ pts, int* __restrict__ vid,
    float* __restrict__ cnt, float* __restrict__ sxyz, float* __restrict__ stats)
{
  __shared__ int bc;
  if (threadIdx.x == 0) bc = 0;
  __syncthreads();
  int i = blockIdx.x * 256 + threadIdx.x;
  if (i < NPTS) {
    float px = pts[3 * i + 0], py = pts[3 * i + 1], pz = pts[3 * i + 2];
    int ix = (int)floorf((px - (-51.2f)) / 0.2f);
    int iy = (int)floorf((py - (-51.2f)) / 0.2f);
    int iz = (int)floorf((pz - (-3.0f)) / 6.0f);
    bool valid = (ix >= 0) && (ix < 512) && (iy >= 0) && (iy < 512) && (iz >= 0) && (iz < 1);
    int v = valid ? (iy * WXD + ix) : HW;
    vid[i] = v;
    if (valid) {
      atomicAdd(&cnt[v], 1.0f);
      atomicAdd(&sxyz[(size_t)v * 3 + 0], px);
      atomicAdd(&sxyz[(size_t)v * 3 + 1], py);
      atomicAdd(&sxyz[(size_t)v * 3 + 2], pz);
      atomicAdd(&bc, 1);
    }
  }
  __syncthreads();
  if (threadIdx.x == 0 && bc > 0) atomicAdd(&stats[IDX_N], (float)bc);
}

/* ============ K2: feats + GEMM1 (9x32) + BN1 stats ============ */
__global__ __launch_bounds__(256) void k2_gemm1(
    const float* __restrict__ pts, const int* __restrict__ vid,
    const float* __restrict__ cnt, const float* __restrict__ sxyz,
    const float* __restrict__ W1, float* __restrict__ y1, float* __restrict__ stats)
{
  __shared__ float w1s[9 * 32];
  __shared__ float ls[32], lq[32];
  int tid = threadIdx.x;
  if (tid < 288) w1s[tid] = W1[tid];
  if (tid < 32) { ls[tid] = 0.0f; lq[tid] = 0.0f; }
  __syncthreads();
  int i = blockIdx.x * 256 + tid;
  if (i < NPTS) {
    int v = vid[i];
    bool valid = v < HW;
    float y[32];
    if (valid) {
      float px = pts[3 * i + 0], py = pts[3 * i + 1], pz = pts[3 * i + 2];
      float inv = 1.0f / fmaxf(cnt[v], 1.0f);
      float mx = sxyz[(size_t)v * 3 + 0] * inv;
      float my = sxyz[(size_t)v * 3 + 1] * inv;
      float mz = sxyz[(size_t)v * 3 + 2] * inv;
      int cx = v & 511, cy = v >> 9;
      float ctx = -51.2f + ((float)cx + 0.5f) * 0.2f;
      float cty = -51.2f + ((float)cy + 0.5f) * 0.2f;
      float ctz = -3.0f + 0.5f * 6.0f;
      float f[9] = {px, py, pz, px - mx, py - my, pz - mz, px - ctx, py - cty, pz - ctz};
      #pragma unroll
      for (int c = 0; c < 32; ++c) {
        float acc = 0.0f;
        #pragma unroll
        for (int k = 0; k < 9; ++k) acc += f[k] * w1s[k * 32 + c];
        y[c] = acc;
      }
      #pragma unroll
      for (int c = 0; c < 32; ++c) {
        atomicAdd(&ls[c], y[c]);
        atomicAdd(&lq[c], y[c] * y[c]);
      }
    } else {
      #pragma unroll
      for (int c = 0; c < 32; ++c) y[c] = 0.0f;
    }
    #pragma unroll
    for (int c = 0; c < 32; ++c) y1[(size_t)i * 32 + c] = y[c];
  }
  __syncthreads();
  if (tid < 32) {
    atomicAdd(&stats[IDX_S1 + tid], ls[tid]);
    atomicAdd(&stats[IDX_S1Q + tid], lq[tid]);
  }
}

/* ============ K3/K7: BN param fold ============ */
__global__ void k_bnparams(float* __restrict__ stats, const float* __restrict__ g,
                           const float* __restrict__ b, int C, int iS, int iQ, int iSC, int iSH)
{
  int c = threadIdx.x;
  if (c < C) {
    float n = fmaxf(stats[IDX_N], 1.0f);
    float mean = stats[iS + c] / n;
    float var = fmaxf(stats[iQ + c] / n - mean * mean, 0.0f);
    float sc = g[c] * rsqrtf(var + EPSB);
    stats[iSC + c] = sc;
    stats[iSH + c] = b[c] - mean * sc;
  }
}

/* ============ K4: ReLU(BN1) + pool1 scatter ============ */
__global__ __launch_bounds__(256) void k4_act1(
    const int* __restrict__ vid, float* __restrict__ x1,
    float* __restrict__ p1sum, float* __restrict__ p1max, const float* __restrict__ stats)
{
  int i = blockIdx.x * 256 + threadIdx.x;
  if (i >= NPTS) return;
  int v = vid[i];
  bool valid = v < HW;
  #pragma unroll
  for (int c = 0; c < 32; ++c) {
    float x = 0.0f;
    if (valid) x = fmaxf(stats[IDX_SC1 + c] * x1[(size_t)i * 32 + c] + stats[IDX_SH1 + c], 0.0f);
    x1[(size_t)i * 32 + c] = x;
    if (valid) {
      atomicAdd(&p1sum[(size_t)v * 32 + c], x);
      atomicMax((unsigned int*)&p1max[(size_t)v * 32 + c], __float_as_uint(x));
    }
  }
}

/* ============ K5: pool1 finalize (in place into p1sum) ============ */
__global__ __launch_bounds__(256) void k5_pool1(
    float* __restrict__ p1sum, const float* __restrict__ p1max, const float* __restrict__ cnt)
{
  size_t t = (size_t)blockIdx.x * 256 + threadIdx.x;
  int v = (int)(t >> 5);
  float s = p1sum[t];
  float m = p1max[t];
  p1sum[t] = 0.5f * (m + s / fmaxf(cnt[v], 1.0f));
}

/* ============ K6: WMMA GEMM2 (Nx64 @ 64x64, f32 16x16x4) + BN2 stats =====
 * LDS W2 is stored K-pair swizzled: w2s[(k>>1)*128 + n*2 + (k&1)] so each
 * lane's B fragment (W2[k][n], W2[k+1][n]) is one aligned ds_load_b64.    */
__global__ __launch_bounds__(128) void k6_gemm2(
    const float* __restrict__ x1, const float* __restrict__ p1,
    const int* __restrict__ vid, const float* __restrict__ W2,
    float* __restrict__ y2, float* __restrict__ stats)
{
  __shared__ float w2s[64 * 64];
  __shared__ float s2s[64], s2q[64];
  int tid = threadIdx.x;
  for (int i = tid; i < 64 * 64; i += 128) {
    int k = i >> 6, n = i & 63;
    w2s[(k >> 1) * 128 + n * 2 + (k & 1)] = W2[i];
  }
  if (tid < 64) { s2s[tid] = 0.0f; s2q[tid] = 0.0f; }
  __syncthreads();

  int wave = tid >> 5;
  int lane = tid & 31;
  int mtile = blockIdx.x * 4 + wave;
  if (mtile * 16 < NPTS) {            /* wave-uniform */
    int m0 = mtile * 16;
    int half = lane >> 4;
    int nb = lane & 15;
    int row = m0 + nb;
    int vr = vid[row];
    bool vld = vr < HW;
    const float* x1row = x1 + (size_t)row * 32;
    const float* p1row = p1 + (size_t)(vld ? vr : 0) * 32;
    v8f acc0 = (v8f)0.0f, acc1 = (v8f)0.0f, acc2 = (v8f)0.0f, acc3 = (v8f)0.0f;

    #pragma unroll
    for (int kc = 0; kc < 16; ++kc) {
      int k = kc * 4 + half * 2;
      float a0, a1;
      if (kc < 8) {                   /* K 0..31: x1 part */
        a0 = x1row[k];
        a1 = x1row[k + 1];
      } else {                        /* K 32..63: pooled part */
        a0 = vld ? p1row[k - 32] : 0.0f;
        a1 = vld ? p1row[k - 31] : 0.0f;
      }
      v2f av; av.x = a0; av.y = a1;
      int kp = (kc * 2 + half) * 128; /* = (k>>1)*128 */
      v2f b0 = *(const v2f*)&w2s[kp + nb * 2];
      acc0 = __builtin_amdgcn_wmma_f32_16x16x4_f32(false, av, false, b0, (short)0, acc0, false, false);
      v2f b1 = *(const v2f*)&w2s[kp + (16 + nb) * 2];
      acc1 = __builtin_amdgcn_wmma_f32_16x16x4_f32(false, av, false, b1, (short)0, acc1, false, false);
      v2f b2 = *(const v2f*)&w2s[kp + (32 + nb) * 2];
      acc2 = __builtin_amdgcn_wmma_f32_16x16x4_f32(false, av, false, b2, (short)0, acc2, false, false);
      v2f b3 = *(const v2f*)&w2s[kp + (48 + nb) * 2];
      acc3 = __builtin_amdgcn_wmma_f32_16x16x4_f32(false, av, false, b3, (short)0, acc3, false, false);
    }

    /* write y2: C/D layout lane<16 -> M=r, lane>=16 -> M=r+8, N=lane&15 */
    #pragma unroll
    for (int r = 0; r < 8; ++r) {
      size_t rw = (size_t)(m0 + r + 8 * half);
      float* yo = y2 + rw * 64 + nb;
      yo[0]  = acc0[r];
      yo[16] = acc1[r];
      yo[32] = acc2[r];
      yo[48] = acc3[r];
    }
    /* BN2 stats partials */
    float s0 = 0, q0 = 0, s1 = 0, q1 = 0, s2 = 0, q2 = 0, s3 = 0, q3 = 0;
    #pragma unroll
    for (int r = 0; r < 8; ++r) {
      s0 += acc0[r]; q0 += acc0[r] * acc0[r];
      s1 += acc1[r]; q1 += acc1[r] * acc1[r];
      s2 += acc2[r]; q2 += acc2[r] * acc2[r];
      s3 += acc3[r]; q3 += acc3[r] * acc3[r];
    }
    atomicAdd(&s2s[nb],      s0); atomicAdd(&s2q[nb],      q0);
    atomicAdd(&s2s[16 + nb], s1); atomicAdd(&s2q[16 + nb], q1);
    atomicAdd(&s2s[32 + nb], s2); atomicAdd(&s2q[32 + nb], q2);
    atomicAdd(&s2s[48 + nb], s3); atomicAdd(&s2q[48 + nb], q3);
  }
  __syncthreads();
  if (tid < 64) {
    atomicAdd(&stats[IDX_S2 + tid], s2s[tid]);
    atomicAdd(&stats[IDX_S2Q + tid], s2q[tid]);
  }
}

/* ============ K8: ReLU(BN2) + pool2 scatter ============ */
__global__ __launch_bounds__(256) void k8_act2(
    const int* __restrict__ vid, const float* __restrict__ y2,
    float* __restrict__ p2sum, float* __restrict__ p2max, const float* __restrict__ stats)
{
  int i = blockIdx.x * 256 + threadIdx.x;
  if (i >= NPTS) return;
  int v = vid[i];
  if (v >= HW) return;                /* invalid rows contribute only to dropped segment */
  #pragma unroll
  for (int c = 0; c < 64; ++c) {
    float x = fmaxf(stats[IDX_SC2 + c] * y2[(size_t)i * 64 + c] + stats[IDX_SH2 + c], 0.0f);
    atomicAdd(&p2sum[(size_t)v * 64 + c], x);
    atomicMax((unsigned int*)&p2max[(size_t)v * 64 + c], __float_as_uint(x));
  }
}

/* ============ K9: pool2 finalize + transpose to (C,H,W) ============ */
__global__ __launch_bounds__(256) void k9_finalize(
    const float* __restrict__ p2sum, const float* __restrict__ p2max,
    const float* __restrict__ cnt, float* __restrict__ out)
{
  __shared__ float t[64 * 65];
  int v0 = blockIdx.x * 64;
  #pragma unroll
  for (int j = 0; j < 16; ++j) {
    int e = threadIdx.x + j * 256;    /* 4096 elements: 64 voxels x 64 ch */
    int vv = e >> 6, c = e & 63;
    size_t idx = (size_t)(v0 + vv) * 64 + c;
    float val = 0.5f * (p2max[idx] + p2sum[idx] / fmaxf(cnt[v0 + vv], 1.0f));
    t[c * 65 + vv] = val;
  }
  __syncthreads();
  #pragma unroll
  for (int j = 0; j < 16; ++j) {
    int e = threadIdx.x + j * 256;
    int c = e >> 6, vv = e & 63;
    out[(size_t)c * HW + v0 + vv] = t[c * 65 + vv];
  }
}

/* =================== host launcher =================== */
extern "C" void kernel_launch(void* const* d_in, const int* in_sizes, int n_in,
                              void* d_out, int out_size, void* d_ws, size_t ws_size,
                              hipStream_t stream) {
  (void)in_sizes; (void)n_in; (void)out_size;
  if (ws_size < WS_NEEDED) return;

  const float* pts_all = (const float*)d_in[0];
  const float* W1 = (const float*)d_in[1];
  const float* g1 = (const float*)d_in[2];
  const float* b1 = (const float*)d_in[3];
  const float* W2 = (const float*)d_in[4];
  const float* g2 = (const float*)d_in[5];
  const float* b2 = (const float*)d_in[6];

  char* ws = (char*)d_ws;
  float* stats = (float*)(ws + O_STATS);
  int*   vid   = (int*)(ws + O_VID);
  float* cnt   = (float*)(ws + O_CNT);
  float* x1b   = (float*)(ws + O_X1);
  float* sxyz  = (float*)(ws + O_SXYZ);
  float* p1sum = (float*)(ws + O_P1SUM);
  float* p1max = (float*)(ws + O_P1MAX);
  float* p2max = (float*)(ws + O_P2MAX);
  float* y2b   = (float*)(ws + O_Y2);
  float* p2sum = (float*)(ws + O_P2SUM);

  const int PB = (NPTS + 255) / 256;           /* point-parallel blocks */
  const int G6 = ((NPTS / 16) + 3) / 4;        /* 9375 M-tiles, 4 waves/block */

  for (int b = 0; b < 2; ++b) {
    const float* pts = pts_all + (size_t)b * NPTS * 3;
    float* out = (float*)d_out + (size_t)b * 64 * HW;

    hipMemsetAsync(ws + O_STATS, 0, SZ_STATS, stream);
    hipMemsetAsync(ws + O_CNT, 0, SZ_CNT, stream);
    hipMemsetAsync(ws + O_SXYZ, 0, SZ_SXYZ, stream);
    hipMemsetAsync(ws + O_P1SUM, 0, 2 * SZ_P1, stream);

    k1_voxelize<<<PB, 256, 0, stream>>>(pts, vid, cnt, sxyz, stats);
    k2_gemm1<<<PB, 256, 0, stream>>>(pts, vid, cnt, sxyz, W1, x1b, stats);
    k_bnparams<<<1, 32, 0, stream>>>(stats, g1, b1, 32, IDX_S1, IDX_S1Q, IDX_SC1, IDX_SH1);
    k4_act1<<<PB, 256, 0, stream>>>(vid, x1b, p1sum, p1max, stats);
    k5_pool1<<<(HW * 32) / 256, 256, 0, stream>>>(p1sum, p1max, cnt);
    k6_gemm2<<<G6, 128, 0, stream>>>(x1b, p1sum, vid, W2, y2b, stats);
    k_bnparams<<<1, 64, 0, stream>>>(stats, g2, b2, 64, IDX_S2, IDX_S2Q, IDX_SC2, IDX_SH2);

    /* p1/x1 region dead now -> safe to overlay p2max */
    hipMemsetAsync(ws + O_P2MAX, 0, SZ_P2, stream);
    hipMemsetAsync(ws + O_P2SUM, 0, SZ_P2, stream);

    k8_act2<<<PB, 256, 0, stream>>>(vid, y2b, p2sum, p2max, stats);
    k9_finalize<<<HW / 64, 256, 0, stream>>>(p2sum, p2max, cnt, out);
  }
}